// SampleCutting_33311766347845
// MI455X (gfx1250) — compile-verified
//
#include <hip/hip_runtime.h>
#include <hip/hip_bf16.h>

// ---------------------------------------------------------------------------
// iSTFT -> sample cut -> STFT, formulated as two f32 WMMA GEMMs against
// precomputed DFT basis matrices (1022 is not an FFT-friendly size; the
// problem is bandwidth-bound: ~67 GFLOP vs ~330 MB HBM traffic -> ~14us at
// 23.3 TB/s, so f32 precision via V_WMMA_F32_16X16X4_F32 is the right call).
// ---------------------------------------------------------------------------

#define WIN   1022
#define HOP   256
#define T_IN  127
#define NB    128
#define LSIG  33278       // HOP*(T_IN-1) + WIN
#define LPAD  33280
#define KEEP  32278       // LSIG - 1000
#define CSTRIDE 32288
#define T_OUT 123
#define NFREQ 512
#define M1    (NB * T_IN)    // 16256
#define M2    (NB * T_OUT)   // 15744
#define PI2   6.28318530717958647692f

typedef __attribute__((ext_vector_type(2))) float v2f;
typedef __attribute__((ext_vector_type(8))) float v8f;

// Workspace layout (floats)
#define OFF_B1   0
#define SZ_B1    (512 * 1024 * 2)              // 1,048,576
#define OFF_B2   (OFF_B1 + SZ_B1)
#define SZ_B2    (512 * 1024 * 2)              // 1,048,576
#define OFF_FR   (OFF_B2 + SZ_B2)
#define SZ_FR    (M1 * 1024)                   // 16,646,144
#define OFF_SIG  (OFF_FR + SZ_FR)
#define SZ_SIG   (NB * LPAD)                   // 4,259,840
#define OFF_CUT  (OFF_SIG + SZ_SIG)
#define SZ_CUT   (NB * CSTRIDE)                // 4,132,864

__device__ __forceinline__ float hann_at(int n) {
    return 0.5f - 0.5f * cosf(PI2 * (float)n * (1.0f / (float)WIN));
}

// basis1[(k*1024 + n)*2 + {0,1}] : K-pair-interleaved inverse-DFT basis,
// row 2k = coef_k/N * cos(2pi*k*n/N) * INV_WIN[n], row 2k+1 = -coef'_k/N * sin * INV_WIN[n]
__global__ void build_basis1(float* __restrict__ basis1) {
    int idx = blockIdx.x * 256 + threadIdx.x;
    if (idx >= 512 * 1024) return;
    int k = idx >> 10;
    int n = idx & 1023;
    float cv = 0.0f, sv = 0.0f;
    if (n < WIN) {
        int kn = (k * n) % WIN;                       // exact integer reduction
        float th = PI2 * (float)kn * (1.0f / (float)WIN);
        float s, c;
        sincosf(th, &s, &c);
        float w = hann_at(n);
        // inverse-STFT window denominator: sum of squared hann over overlaps
        int j = n & 255;
        float d = 0.0f;
        #pragma unroll
        for (int i = 0; i < 4; ++i) {
            int p = i * 256 + j;
            if (p < WIN) { float wp = hann_at(p); d += wp * wp; }
        }
        float scale = (w / d) * (1.0f / (float)WIN);
        float ck = (k == 0 || k == 511) ? 1.0f : 2.0f;
        float sk = (k == 0 || k == 511) ? 0.0f : 2.0f;
        cv =  ck * scale * c;
        sv = -sk * scale * s;
    }
    basis1[(size_t)idx * 2 + 0] = cv;
    basis1[(size_t)idx * 2 + 1] = sv;
}

// basis2[(s*1024 + n)*2 + c2] : forward-DFT basis, row index kI = 2s + c2
// (sample), col n = 2k' + cc (re/im of bin k'), Hann window folded in.
// Rows kI >= 1022 are zero so K can be padded to 1024.
__global__ void build_basis2(float* __restrict__ basis2) {
    int idx = blockIdx.x * 256 + threadIdx.x;
    if (idx >= 512 * 1024) return;
    int s = idx >> 10;
    int n = idx & 1023;
    int kp = n >> 1;
    int cc = n & 1;
    #pragma unroll
    for (int c2 = 0; c2 < 2; ++c2) {
        int kI = 2 * s + c2;
        float v = 0.0f;
        if (kI < WIN) {
            float w = hann_at(kI);
            int r = (kp * kI) % WIN;
            float th = PI2 * (float)r * (1.0f / (float)WIN);
            float sn, cs;
            sincosf(th, &sn, &cs);
            v = (cc == 0) ? (w * cs) : (-w * sn);
        }
        basis2[(size_t)idx * 2 + c2] = v;
    }
}

// GEMM1: frames[m][n] = sum_kk A[m][kk] * basis1row[kk][n]
// A[m=(b*127+t)][2k+c] = in[((b*512+k)*127+t)*2+c]; one wave = 16x64 tile.
__global__ void gemm_irfft(const float* __restrict__ in,
                           const float* __restrict__ basis1,
                           float* __restrict__ frames) {
    const int lane = threadIdx.x & 31;
    const int wave = threadIdx.x >> 5;
    const int m0   = blockIdx.x * 16;
    const int n0   = (blockIdx.y * 4 + wave) * 64;
    const int half = lane >> 4;
    const int col  = lane & 15;
    const int mrow = m0 + col;
    const int b    = mrow / T_IN;
    const int t    = mrow - b * T_IN;
    const long abase = (long)b * (NFREQ * T_IN * 2) + t * 2;   // + k*254

    v8f a0 = {}, a1 = {}, a2 = {}, a3 = {};
    for (int i = 0; i < 256; ++i) {
        const int s = 2 * i + half;      // lane's frequency index (A) / row pair (B)
        v2f av = *(const v2f*)(in + abase + (long)s * (T_IN * 2));
        const float* brow = basis1 + ((long)s * 1024 + n0 + col) * 2;
        v2f b0 = *(const v2f*)(brow + 0);
        v2f b1 = *(const v2f*)(brow + 32);
        v2f b2 = *(const v2f*)(brow + 64);
        v2f b3 = *(const v2f*)(brow + 96);
        a0 = __builtin_amdgcn_wmma_f32_16x16x4_f32(false, av, false, b0, (short)0, a0, false, false);
        a1 = __builtin_amdgcn_wmma_f32_16x16x4_f32(false, av, false, b1, (short)0, a1, false, false);
        a2 = __builtin_amdgcn_wmma_f32_16x16x4_f32(false, av, false, b2, (short)0, a2, false, false);
        a3 = __builtin_amdgcn_wmma_f32_16x16x4_f32(false, av, false, b3, (short)0, a3, false, false);
    }
    // C layout: VGPR v -> row m0 + v + 8*half, col = n0 + j*16 + (lane&15)
    #pragma unroll
    for (int v = 0; v < 8; ++v) {
        long r = (long)(m0 + v + 8 * half) * 1024 + n0 + col;
        frames[r +  0] = a0[v];
        frames[r + 16] = a1[v];
        frames[r + 32] = a2[v];
        frames[r + 48] = a3[v];
    }
}

// Deterministic overlap-add as a gather (<= 4 frames cover each sample).
__global__ void overlap_add(const float* __restrict__ frames,
                            float* __restrict__ signals) {
    long idx = (long)blockIdx.x * 256 + threadIdx.x;
    if (idx >= (long)NB * LSIG) return;
    int b = (int)(idx / LSIG);
    int p = (int)(idx - (long)b * LSIG);
    int tmin = (p > WIN - 1) ? ((p - (WIN - 1) + HOP - 1) >> 8) : 0;
    int tmax = p >> 8;
    if (tmax > T_IN - 1) tmax = T_IN - 1;
    float s = 0.0f;
    for (int t = tmin; t <= tmax; ++t)
        s += frames[(long)(b * T_IN + t) * 1024 + (p - t * HOP)];
    signals[(long)b * LPAD + p] = s;
}

// Sample cut: gather keep_idx.
__global__ void cut_gather(const float* __restrict__ signals,
                           const int* __restrict__ keep_idx,
                           float* __restrict__ cutsig) {
    long idx = (long)blockIdx.x * 256 + threadIdx.x;
    if (idx >= (long)NB * KEEP) return;
    int b = (int)(idx / KEEP);
    int j = (int)(idx - (long)b * KEEP);
    cutsig[(long)b * CSTRIDE + j] = signals[(long)b * LPAD + keep_idx[j]];
}

// GEMM2: out_bins = frames_cut x basis2, window folded into basis2;
// K padded to 1024 with zero basis rows. One wave = 16x64 tile.
__global__ void gemm_rfft(const float* __restrict__ cutsig,
                          const float* __restrict__ basis2,
                          float* __restrict__ out) {
    const int lane = threadIdx.x & 31;
    const int wave = threadIdx.x >> 5;
    const int m0   = blockIdx.x * 16;
    const int n0   = (blockIdx.y * 4 + wave) * 64;
    const int half = lane >> 4;
    const int col  = lane & 15;
    const int mrow = m0 + col;
    const int b    = mrow / T_OUT;
    const int t    = mrow - b * T_OUT;
    const long abase = (long)b * CSTRIDE + t * HOP;

    v8f a0 = {}, a1 = {}, a2 = {}, a3 = {};
    for (int i = 0; i < 256; ++i) {
        const int s = 2 * i + half;                    // sample-pair index
        v2f av = *(const v2f*)(cutsig + abase + 2 * s); // samples kI=2s, 2s+1
        const float* brow = basis2 + ((long)s * 1024 + n0 + col) * 2;
        v2f b0 = *(const v2f*)(brow + 0);
        v2f b1 = *(const v2f*)(brow + 32);
        v2f b2 = *(const v2f*)(brow + 64);
        v2f b3 = *(const v2f*)(brow + 96);
        a0 = __builtin_amdgcn_wmma_f32_16x16x4_f32(false, av, false, b0, (short)0, a0, false, false);
        a1 = __builtin_amdgcn_wmma_f32_16x16x4_f32(false, av, false, b1, (short)0, a1, false, false);
        a2 = __builtin_amdgcn_wmma_f32_16x16x4_f32(false, av, false, b2, (short)0, a2, false, false);
        a3 = __builtin_amdgcn_wmma_f32_16x16x4_f32(false, av, false, b3, (short)0, a3, false, false);
    }
    // out[b, k'=n>>1, t, c=n&1], out shape (NB, 512, 123, 2)
    #pragma unroll
    for (int v = 0; v < 8; ++v) {
        int m = m0 + v + 8 * half;
        int bb = m / T_OUT;
        int tt = m - bb * T_OUT;
        #pragma unroll
        for (int j = 0; j < 4; ++j) {
            int n = n0 + j * 16 + col;
            float val = (j == 0) ? a0[v] : (j == 1) ? a1[v] : (j == 2) ? a2[v] : a3[v];
            out[(((long)bb * NFREQ + (n >> 1)) * T_OUT + tt) * 2 + (n & 1)] = val;
        }
    }
}

extern "C" void kernel_launch(void* const* d_in, const int* in_sizes, int n_in,
                              void* d_out, int out_size, void* d_ws, size_t ws_size,
                              hipStream_t stream) {
    const float* in      = (const float*)d_in[0];
    const int*  keep_idx = (const int*)d_in[1];
    float* out = (float*)d_out;
    float* ws  = (float*)d_ws;

    float* basis1  = ws + OFF_B1;
    float* basis2  = ws + OFF_B2;
    float* frames  = ws + OFF_FR;
    float* signals = ws + OFF_SIG;
    float* cutsig  = ws + OFF_CUT;

    build_basis1<<<(512 * 1024) / 256, 256, 0, stream>>>(basis1);
    build_basis2<<<(512 * 1024) / 256, 256, 0, stream>>>(basis2);

    gemm_irfft<<<dim3(M1 / 16, 4), 128, 0, stream>>>(in, basis1, frames);

    long nsig = (long)NB * LSIG;
    overlap_add<<<(int)((nsig + 255) / 256), 256, 0, stream>>>(frames, signals);

    long ncut = (long)NB * KEEP;
    cut_gather<<<(int)((ncut + 255) / 256), 256, 0, stream>>>(signals, keep_idx, cutsig);

    gemm_rfft<<<dim3(M2 / 16, 4), 128, 0, stream>>>(cutsig, basis2, out);

    (void)in_sizes; (void)n_in; (void)out_size; (void)ws_size;
}